// BoundaryLoss_11596411699808
// MI455X (gfx1250) — compile-verified
//
#include <hip/hip_runtime.h>
#include <stdint.h>

// Problem constants (B,C,H,W) = (4,8,192,192)
#define NB 4
#define NC 8
#define NH 192
#define NW 192
#define NTILE 16
#define NCT   (NW / NTILE)        // 12 column tiles
#define NBLK3 (NB * NC * NCT)     // 384 blocks in column pass (divisible by 64)
#define NPIX  (NB * NH * NW)      // 147456
#define NG    (NB * NC * NH * NW) // 1179648
#define BIG2F 1.0e12f             // fl(BIG^2), matches reference fp32 g*g
#define MBIGF 4.0e12f             // "not in set" additive sentinel (> BIG2 cap)

typedef __attribute__((ext_vector_type(2))) float v2f;
typedef __attribute__((ext_vector_type(8))) float v8f;

// ---------------------------------------------------------------- helpers
__device__ __forceinline__ uint32_t lds_lo32(const void* p) {
  // addrspacecast(LDS->flat) keeps the LDS byte offset in the low 32 bits
  return (uint32_t)(uintptr_t)p;
}

// CDNA5 async DMA: global -> LDS, 4B per lane, GVS addressing (saddr + vgpr off)
__device__ __forceinline__ void async_g2l_b32(uint32_t lds_byte, const void* sbase,
                                              uint32_t voff) {
  asm volatile("global_load_async_to_lds_b32 %0, %1, %2 offset:0"
               :: "v"(lds_byte), "v"(voff), "s"(sbase) : "memory");
}
__device__ __forceinline__ void wait_async0() {
  asm volatile("s_wait_asynccnt 0" ::: "memory");
}

// wave32 butterfly reductions
__device__ __forceinline__ float wred_f(float v) {
#pragma unroll
  for (int o = 16; o > 0; o >>= 1) v += __shfl_xor(v, o, 32);
  return v;
}
__device__ __forceinline__ int wred_i(int v) {
#pragma unroll
  for (int o = 16; o > 0; o >>= 1) v += __shfl_xor(v, o, 32);
  return v;
}

// ---------------------------------------------------------------- kernels
__global__ __launch_bounds__(256) void k_init(int* flags, float* part) {
  int t = threadIdx.x;
  if (t < NB * NC) flags[t] = 0;
  for (int i = t; i < NBLK3; i += 256) part[i] = 0.0f;
}

// Per-pixel softmax stats over C=8 (+ has_pos flags)
__global__ __launch_bounds__(256) void k_stats(const float* __restrict__ x,
                                               const int* __restrict__ tg,
                                               float* __restrict__ mx,
                                               float* __restrict__ sei,
                                               int* __restrict__ flags) {
  int p = blockIdx.x * 256 + threadIdx.x;
  if (p >= NPIX) return;
  __builtin_prefetch(tg + p, 0, 0);  // global_prefetch_b8
  int b = p / (NH * NW);
  int ij = p - b * (NH * NW);
  const float* xb = x + (size_t)b * NC * NH * NW + ij;
  float v[NC];
  float m = -3.4e38f;
#pragma unroll
  for (int c = 0; c < NC; ++c) {
    v[c] = xb[(size_t)c * NH * NW];
    m = fmaxf(m, v[c]);
  }
  float s = 0.0f;
#pragma unroll
  for (int c = 0; c < NC; ++c) s += expf(v[c] - m);
  mx[p] = m;
  sei[p] = 1.0f / s;
  int t = tg[p];
  bool valid = (t != -100) && (t >= 0) && (t < NC);
  if (valid) atomicOr(&flags[b * NC + t], 1);  // idempotent -> deterministic
}

// Row-axis EDT (both masks), min-plus form: g2 = min(BIG2, min_k m2[k]+(j-k)^2)
__global__ __launch_bounds__(192) void k_rowedt(const int* __restrict__ tg,
                                                float* __restrict__ g2o,
                                                float* __restrict__ g2i) {
  __shared__ int   ldsT[NW];
  __shared__ float m2o[NW];
  __shared__ float m2i[NW];
  int idx = blockIdx.x;          // (b,c,i)
  int i  = idx % NH;
  int bc = idx / NH;
  int b  = bc / NC;
  int c  = bc % NC;
  int j  = threadIdx.x;

  const int* rowBase = tg + (size_t)b * NH * NW + (size_t)i * NW;
  async_g2l_b32(lds_lo32(&ldsT[j]), rowBase, (uint32_t)(j * 4));
  wait_async0();
  __syncthreads();

  int t = ldsT[j];
  bool valid = (t != -100) && (t >= 0) && (t < NC);
  bool msk = valid && (t == c);
  m2o[j] = msk ? 0.0f : MBIGF;   // dist-to-class-foreground
  m2i[j] = msk ? MBIGF : 0.0f;   // dist-to-complement
  __syncthreads();

  float d1 = BIG2F, d2 = BIG2F;  // init with cap == min(..., BIG)^2
  float df = (float)j;
#pragma unroll 4
  for (int k = 0; k < NW; ++k) {
    d1 = fminf(d1, fmaf(df, df, m2o[k]));
    d2 = fminf(d2, fmaf(df, df, m2i[k]));
    df -= 1.0f;
  }
  size_t o = ((size_t)bc * NH + i) * NW + j;
  g2o[o] = d1;
  g2i[o] = d2;
}

// Column-axis EDT + fused loss accumulation over a 16-column tile of one (b,c)
__global__ __launch_bounds__(256) void k_coledt(const float* __restrict__ g2o,
                                                const float* __restrict__ g2i,
                                                const float* __restrict__ x,
                                                const int* __restrict__ tg,
                                                const float* __restrict__ mx,
                                                const float* __restrict__ sei,
                                                const int* __restrict__ flags,
                                                float* __restrict__ part) {
  __shared__ float tO[NH * NTILE];
  __shared__ float tI[NH * NTILE];
  __shared__ float red[8];
  int blk = blockIdx.x;
  int ct  = blk % NCT;
  int bc  = blk / NCT;
  int b   = bc / NC;
  int colBase = ct * NTILE;
  int tid = threadIdx.x;

  // Async-DMA both g^2 tiles (H x 16) straight into LDS
  const float* baseO = g2o + (size_t)bc * NH * NW + colBase;
  const float* baseI = g2i + (size_t)bc * NH * NW + colBase;
  for (int e = tid; e < NH * NTILE; e += 256) {
    int k = e >> 4, j = e & 15;
    uint32_t voff = (uint32_t)((k * NW + j) * 4);
    async_g2l_b32(lds_lo32(&tO[e]), baseO, voff);
    async_g2l_b32(lds_lo32(&tI[e]), baseI, voff);
  }
  wait_async0();
  __syncthreads();

  int j  = tid & 15;
  int i0 = tid >> 4;
  bool hp = (flags[bc] != 0);
  float acc = 0.0f;
  size_t planeBase = (size_t)bc * NH * NW + colBase + j;  // x index base (b,c)
  size_t pixBase   = (size_t)b  * NH * NW + colBase + j;  // per-pixel (b)

#pragma unroll 1
  for (int r = 0; r < NH / 16; ++r) {
    int i = i0 + r * 16;
    float d2o = 3.4e38f, d2i = 3.4e38f;
    float df = (float)i;
#pragma unroll 4
    for (int k = 0; k < NH; ++k) {
      float go = tO[(k << 4) + j];
      float gi = tI[(k << 4) + j];
      d2o = fminf(d2o, fmaf(df, df, go));
      d2i = fminf(d2i, fmaf(df, df, gi));
      df -= 1.0f;
    }
    size_t pix = pixBase + (size_t)i * NW;
    int t = tg[pix];
    bool valid = (t != -100) && (t >= 0) && (t < NC);
    if (valid) {
      float s = hp ? (sqrtf(d2o) - sqrtf(d2i)) : 0.0f;
      float prob = expf(x[planeBase + (size_t)i * NW] - mx[pix]) * sei[pix];
      acc += prob * s;
    }
  }

  // fixed-order block reduction (deterministic)
  acc = wred_f(acc);
  int lane = tid & 31, wid = tid >> 5;
  if (lane == 0) red[wid] = acc;
  __syncthreads();
  if (tid < 32) {
    float v = (tid < 8) ? red[tid] : 0.0f;
    v = wred_f(v);
    if (tid == 0) part[blockIdx.x] = v;
  }
}

// Final: exact f32 WMMA reduction of block partials (total = ones^T * P),
// B = all-ones so the A-packing is layout-independent; C accumulates.
__global__ __launch_bounds__(256) void k_final(const float* __restrict__ part,
                                               const int* __restrict__ tg,
                                               float* __restrict__ out) {
  __shared__ float sumW;
  __shared__ int   redi[8];
  int tid = threadIdx.x;

  // denom = count of valid pixels (all 8 waves)
  int cnt = 0;
  for (int p = tid; p < NPIX; p += 256) {
    int t = tg[p];
    cnt += ((t != -100) && (t >= 0) && (t < NC)) ? 1 : 0;
  }
  cnt = wred_i(cnt);
  int lane = tid & 31, wid = tid >> 5;
  if (lane == 0) redi[wid] = cnt;

  // wave 0 (fully active -> EXEC all-1s) reduces 384 partials on the matrix pipe
  if (tid < 32) {
    v2f bones;
    bones.x = 1.0f; bones.y = 1.0f;
    v8f c = {0.0f, 0.0f, 0.0f, 0.0f, 0.0f, 0.0f, 0.0f, 0.0f};
#pragma unroll
    for (int it = 0; it < NBLK3 / 64; ++it) {  // 6 WMMA issues, 64 values each
      v2f a;
      a.x = part[it * 64 + tid * 2 + 0];
      a.y = part[it * 64 + tid * 2 + 1];
      // D = A x ones + C : D[i,j] += rowsum(A)[i]
      c = __builtin_amdgcn_wmma_f32_16x16x4_f32(false, a, false, bones,
                                                (short)0, c, false, false);
    }
    // lane l<16 holds D[0..7, l]; lane l+16 holds D[8..15, l]
    float s = c[0] + c[1] + c[2] + c[3] + c[4] + c[5] + c[6] + c[7];
    s += __shfl_xor(s, 16, 32);  // total = sum_m D[m, 0], exact
    if (tid == 0) sumW = s;
  }
  __syncthreads();

  if (tid == 0) {
    int ctot = redi[0] + redi[1] + redi[2] + redi[3] +
               redi[4] + redi[5] + redi[6] + redi[7];
    out[0] = sumW / ((float)ctot + 1e-8f);
  }
}

// ---------------------------------------------------------------- launcher
extern "C" void kernel_launch(void* const* d_in, const int* in_sizes, int n_in,
                              void* d_out, int out_size, void* d_ws, size_t ws_size,
                              hipStream_t stream) {
  (void)in_sizes; (void)n_in; (void)out_size; (void)ws_size;
  const float* x  = (const float*)d_in[0];
  const int*   tg = (const int*)d_in[1];
  float* out = (float*)d_out;

  float* ws   = (float*)d_ws;               // ws layout (floats):
  float* g2o  = ws;                         // [0, NG)
  float* g2i  = ws + (size_t)NG;            // [NG, 2NG)
  float* mx   = ws + (size_t)2 * NG;        // [2NG, 2NG+NPIX)
  float* sei  = mx + (size_t)NPIX;          // [.., +NPIX)
  int*   flg  = (int*)(sei + (size_t)NPIX); // 32 ints
  float* part = (float*)(flg + 32);         // NBLK3 floats

  k_init  <<<1, 256, 0, stream>>>(flg, part);
  k_stats <<<NPIX / 256, 256, 0, stream>>>(x, tg, mx, sei, flg);
  k_rowedt<<<NB * NC * NH, 192, 0, stream>>>(tg, g2o, g2i);
  k_coledt<<<NBLK3, 256, 0, stream>>>(g2o, g2i, x, tg, mx, sei, flg, part);
  k_final <<<1, 256, 0, stream>>>(part, tg, out);
}